// ChebyshevConvolutionLinSkin_36627481100819
// MI455X (gfx1250) — compile-verified
//
#include <hip/hip_runtime.h>
#include <hip/hip_bf16.h>

typedef float v2f __attribute__((ext_vector_type(2)));
typedef float v8f __attribute__((ext_vector_type(8)));

// ---------------------------------------------------------------- utilities

__global__ void zero_kernel(float* __restrict__ p, long n) {
  long i = (long)blockIdx.x * blockDim.x + threadIdx.x;
  if (i < n) p[i] = 0.0f;
}

// deg[s] += 1 for non-self-loop edges
__global__ void deg_kernel(const int* __restrict__ src, const int* __restrict__ dst,
                           float* __restrict__ deg, int E) {
  int e = blockIdx.x * blockDim.x + threadIdx.x;
  if (e >= E) return;
  int s = src[e], d = dst[e];
  if (s != d) atomicAdd(&deg[s], 1.0f);
}

// in-place deg -> dinv
__global__ void dinv_kernel(float* __restrict__ deg, int N) {
  int i = blockIdx.x * blockDim.x + threadIdx.x;
  if (i >= N) return;
  float v = deg[i];
  deg[i] = (v > 0.0f) ? rsqrtf(v) : 0.0f;
}

// w[e] = -dinv[s] * dinv[d] (0 for self loops)
__global__ void edgew_kernel(const int* __restrict__ src, const int* __restrict__ dst,
                             const float* __restrict__ dinv, float* __restrict__ w, int E) {
  int e = blockIdx.x * blockDim.x + threadIdx.x;
  if (e >= E) return;
  int s = src[e], d = dst[e];
  w[e] = (s != d) ? (-dinv[s] * dinv[d]) : 0.0f;
}

// Xp[r, 0:FP] = { x[r, 0:F], zeros }   (stride padding F -> FP)
__global__ void pad_kernel(const float* __restrict__ x, float* __restrict__ Xp,
                           int F, int FP, long n /* N*FP */) {
  long i = (long)blockIdx.x * blockDim.x + threadIdx.x;
  if (i >= n) return;
  long r = i / FP;
  int  c = (int)(i - r * FP);
  Xp[i] = (c < F) ? x[r * F + c] : 0.0f;
}

// Wt[m][n][k] = W[m][k][n], zero-padded for k in [F, FP)
__global__ void wtrans_kernel(const float* __restrict__ W, float* __restrict__ Wt,
                              int F, int H, int FP, long n /* 3*H*FP */) {
  long i = (long)blockIdx.x * blockDim.x + threadIdx.x;
  if (i >= n) return;
  int m   = (int)(i / ((long)H * FP));
  long rm = i - (long)m * H * FP;
  int nn  = (int)(rm / FP);
  int k   = (int)(rm - (long)nn * FP);
  Wt[i] = (k < F) ? W[(size_t)m * F * H + (size_t)k * H + nn] : 0.0f;
}

// out[dst] += w[e] * h[src] -- one thread per (edge, float4 chunk); width FS is
// padded so no per-element bounds checks are needed (pad columns are zero).
__global__ void prop_kernel(const int* __restrict__ src, const int* __restrict__ dst,
                            const float* __restrict__ w, const float* __restrict__ h,
                            float* __restrict__ out, int FS, int chunks, long total) {
  long idx = (long)blockIdx.x * blockDim.x + threadIdx.x;
  if (idx >= total) return;
  int e = (int)(idx / chunks);
  int c = (int)(idx - (long)e * chunks) * 4;
  float we = w[e];
  if (we == 0.0f) return;
  int s = src[e], d = dst[e];
  const float4 hv = *(const float4*)(h + (size_t)s * FS + c);
  float* op = out + (size_t)d * FS + c;
  atomicAdd(op + 0, we * hv.x);
  atomicAdd(op + 1, we * hv.y);
  atomicAdd(op + 2, we * hv.z);
  atomicAdd(op + 3, we * hv.w);
}

// P = 2*P - X  (Tx2 = 2*prop(Tx1) - Tx0), elementwise over padded buffers
__global__ void cheb2_kernel(float* __restrict__ P, const float* __restrict__ X, long n) {
  long i = (long)blockIdx.x * blockDim.x + threadIdx.x;
  if (i < n) P[i] = 2.0f * P[i] - X[i];
}

// ---------------------------------------------------------------- WMMA GEMM
// out[N,H] = post( A0@W0 + A1@W1 + A2@W2 + bias (+resid) ), H == 128.
// A* have row stride FP (multiple of 4, zero-padded); Wt is [3][H][FP]
// (transposed, padded). 256 threads = 8 waves; wave w owns column tile w.
// The block stages the 3 x 16 x FP A-tile in LDS once; the k-loop is then
// 1 ds_load_b64 (A frag) + 1 global_load_b64 (B frag) + 1 v_wmma_f32_16x16x4_f32.
template <int FP>
__global__ void cheb_gemm_kernel(const float* __restrict__ A0,
                                 const float* __restrict__ A1,
                                 const float* __restrict__ A2,
                                 const float* __restrict__ Wt,   // [3, H, FP]
                                 const float* __restrict__ bias,
                                 const float* __restrict__ resid, // nullable
                                 float* __restrict__ out,
                                 int H, int do_relu) {
  __shared__ float As[3][16][FP];

  const int tid  = threadIdx.x;
  const int lane = tid & 31;
  const int wv   = tid >> 5;           // 0..7 -> column tile
  const int r0   = blockIdx.x << 4;    // row tile base
  const int half = lane >> 4;          // 0: K=k..k+1, 1: K=k+2..k+3
  const int l15  = lane & 15;
  const int col  = (wv << 4) + l15;

  // ---- cooperative stage of the three 16 x FP A tiles (float4 chunks)
  constexpr int CH = FP / 4;
  const float* Amats[3] = {A0, A1, A2};
  for (int idx = tid; idx < 3 * 16 * CH; idx += 256) {
    int m   = idx / (16 * CH);
    int rem = idx - m * (16 * CH);
    int r   = rem / CH;
    int c   = (rem - r * CH) * 4;
    float4 v = *(const float4*)(Amats[m] + (size_t)(r0 + r) * FP + c);
    *(float4*)(&As[m][r][c]) = v;
  }
  __syncthreads();

  // ---- WMMA main loop (no predication: FP % 4 == 0, pads are zero)
  v8f acc = {};
#pragma unroll
  for (int m = 0; m < 3; ++m) {
    const float* bt = Wt + ((size_t)m * H + col) * FP;  // B col fragment base
#pragma unroll 4
    for (int k = 0; k < FP; k += 4) {
      int kk = k + 2 * half;
      v2f a = *(const v2f*)(&As[m][l15][kk]);
      v2f b = *(const v2f*)(bt + kk);
      acc = __builtin_amdgcn_wmma_f32_16x16x4_f32(
          false, a, false, b, (short)0, acc, false, false);
    }
  }

  // ---- epilogue: bias (+resid) (+relu)
  const float bc = bias[col];
#pragma unroll
  for (int i = 0; i < 8; ++i) {
    int row = r0 + i + half * 8;  // C/D layout: lanes 0-15 -> M=i, 16-31 -> M=8+i
    float v = acc[i] + bc;
    if (resid) v += resid[(size_t)row * H + col];
    if (do_relu) v = fmaxf(v, 0.0f);
    out[(size_t)row * H + col] = v;
  }
}

// ---------------------------------------------------------------- head
// logits = log_softmax(h @ Wl + bl), C = 2
__global__ void head_kernel(const float* __restrict__ h, const float* __restrict__ Wl,
                            const float* __restrict__ bl, float* __restrict__ out,
                            int N, int H) {
  int i = blockIdx.x * blockDim.x + threadIdx.x;
  if (i >= N) return;
  float z0 = bl[0], z1 = bl[1];
  const float* hp = h + (size_t)i * H;
  for (int j = 0; j < H; ++j) {
    float hv = hp[j];
    z0 += hv * Wl[2 * j];
    z1 += hv * Wl[2 * j + 1];
  }
  float m = fmaxf(z0, z1);
  float lse = m + logf(expf(z0 - m) + expf(z1 - m));
  out[2 * i]     = z0 - lse;
  out[2 * i + 1] = z1 - lse;
}

// ---------------------------------------------------------------- launch

extern "C" void kernel_launch(void* const* d_in, const int* in_sizes, int n_in,
                              void* d_out, int out_size, void* d_ws, size_t ws_size,
                              hipStream_t stream) {
  const float* x  = (const float*)d_in[0];
  const int*   ei = (const int*)d_in[1];
  const float* W1 = (const float*)d_in[2];
  const float* b1 = (const float*)d_in[3];
  const float* W2 = (const float*)d_in[4];
  const float* b2 = (const float*)d_in[5];
  const float* Wl = (const float*)d_in[6];
  const float* bl = (const float*)d_in[7];

  const int F = 165, H = 128, C = 2;
  const int FP1 = 168;           // F padded to multiple of 4
  const int FP2 = 128;           // layer-2 width (already multiple of 4)
  const int E = in_sizes[1] / 2;
  const int N = in_sizes[0] / F;
  const int* src = ei;
  const int* dst = ei + E;

  // workspace layout (floats; all region sizes are multiples of 4 -> 16B align)
  float* f   = (float*)d_ws;
  float* deg = f;                           // N        (becomes dinv)
  float* w   = deg + N;                     // E
  float* Xp  = w + E;                       // N*FP1    (padded x; reused as h2)
  float* A1  = Xp + (size_t)N * FP1;        // N*FP1    (Tx1 L1; reused as Tx1 L2)
  float* A2  = A1 + (size_t)N * FP1;        // N*FP1    (Tx2 L1; reused as Tx2 L2)
  float* h1  = A2 + (size_t)N * FP1;        // N*H
  float* Wt  = h1 + (size_t)N * H;          // 3*H*FP1  (transposed weights, reused)
  float* B1  = A1;                          // layer-2 reuse (N*H <= N*FP1)
  float* B2  = A2;
  float* h2  = Xp;                          // Xp dead after GEMM1

  const int T = 256;
  auto blocks = [](long n, int t) { return (unsigned)((n + t - 1) / t); };

  // ---- edge normalization
  zero_kernel<<<blocks(N, T), T, 0, stream>>>(deg, N);
  deg_kernel<<<blocks(E, T), T, 0, stream>>>(src, dst, deg, E);
  dinv_kernel<<<blocks(N, T), T, 0, stream>>>(deg, N);
  edgew_kernel<<<blocks(E, T), T, 0, stream>>>(src, dst, deg, w, E);

  // ---- layer 1 (padded width FP1 = 168)
  const long nf1 = (long)N * FP1;
  pad_kernel<<<blocks(nf1, T), T, 0, stream>>>(x, Xp, F, FP1, nf1);
  const long wt1 = 3L * H * FP1;
  wtrans_kernel<<<blocks(wt1, T), T, 0, stream>>>(W1, Wt, F, H, FP1, wt1);

  const int  ch1  = FP1 / 4;
  const long tot1 = (long)E * ch1;
  zero_kernel<<<blocks(nf1, T), T, 0, stream>>>(A1, nf1);
  prop_kernel<<<blocks(tot1, T), T, 0, stream>>>(src, dst, w, Xp, A1, FP1, ch1, tot1);  // Tx1
  zero_kernel<<<blocks(nf1, T), T, 0, stream>>>(A2, nf1);
  prop_kernel<<<blocks(tot1, T), T, 0, stream>>>(src, dst, w, A1, A2, FP1, ch1, tot1);  // L Tx1
  cheb2_kernel<<<blocks(nf1, T), T, 0, stream>>>(A2, Xp, nf1);                          // Tx2
  cheb_gemm_kernel<168><<<N / 16, 256, 0, stream>>>(Xp, A1, A2, Wt, b1, nullptr, h1, H, 1);

  // ---- layer 2 (width 128), residual skip
  const long nh = (long)N * H;
  const long wt2 = 3L * H * FP2;
  wtrans_kernel<<<blocks(wt2, T), T, 0, stream>>>(W2, Wt, H, H, FP2, wt2);

  const int  ch2  = FP2 / 4;
  const long tot2 = (long)E * ch2;
  zero_kernel<<<blocks(nh, T), T, 0, stream>>>(B1, nh);
  prop_kernel<<<blocks(tot2, T), T, 0, stream>>>(src, dst, w, h1, B1, FP2, ch2, tot2);  // Tx1
  zero_kernel<<<blocks(nh, T), T, 0, stream>>>(B2, nh);
  prop_kernel<<<blocks(tot2, T), T, 0, stream>>>(src, dst, w, B1, B2, FP2, ch2, tot2);  // L Tx1
  cheb2_kernel<<<blocks(nh, T), T, 0, stream>>>(B2, h1, nh);                            // Tx2
  cheb_gemm_kernel<128><<<N / 16, 256, 0, stream>>>(h1, B1, B2, Wt, b2, h1, h2, H, 0);

  // ---- head: logits into d_out
  head_kernel<<<blocks(N, T), T, 0, stream>>>(h2, Wl, bl, (float*)d_out, N, H);

  // ---- edge_index passthrough (second tuple element), raw bytes after logits
  long tail_elems = (long)out_size - (long)N * C;
  if (tail_elems > 0) {
    long want = tail_elems * 4;
    long have = (long)in_sizes[1] * 4;
    hipMemcpyAsync((char*)d_out + (size_t)N * C * 4, d_in[1],
                   (size_t)(want < have ? want : have),
                   hipMemcpyDeviceToDevice, stream);
  }
}